// NGCF_62801011802128
// MI455X (gfx1250) — compile-verified
//
#include <hip/hip_runtime.h>

// ---------------------------------------------------------------------------
// NGCF 3-layer propagation for MI455X (gfx1250, wave32, WMMA).
//
// Algebraic refactor:  msg aggregation is linear in x_src, and the Hadamard
// term factors through the per-dst x_i, so each layer is:
//     s1   = segment_sum(norm * x[src], dst)              (edge phase)
//     h    = act( (x + s1) @ W1  +  (x .* s1) @ W2 )      (node GEMM phase)
// Edge phase is HBM/atomic bound (~1.6 GB/layer); GEMM is ~1.6 GFLOP/layer,
// done in full fp32 via V_WMMA_F32_16X16X4_F32 (K=4, 16 steps over K=64).
// ---------------------------------------------------------------------------

typedef float v2f __attribute__((ext_vector_type(2)));
typedef float v8f __attribute__((ext_vector_type(8)));

#define DFEAT 64
#define NEG_SLOPE 0.01f

// ---- degree count: deg[i] = #edges with dst==i (as float) -----------------
__global__ void ngcf_deg_kernel(const int* __restrict__ dst,
                                float* __restrict__ deg, int n_edges) {
    int e = blockIdx.x * blockDim.x + threadIdx.x;
    if (e < n_edges) atomicAdd(&deg[dst[e]], 1.0f);
}

// ---- per-edge norm = rsqrt(max(deg_s,1)*max(deg_d,1)) ---------------------
__global__ void ngcf_norm_kernel(const int* __restrict__ src,
                                 const int* __restrict__ dst,
                                 const float* __restrict__ deg,
                                 float* __restrict__ nrm, int n_edges) {
    int e = blockIdx.x * blockDim.x + threadIdx.x;
    if (e < n_edges) {
        float ds = fmaxf(deg[src[e]], 1.0f);
        float dd = fmaxf(deg[dst[e]], 1.0f);
        nrm[e] = rsqrtf(ds * dd);
    }
}

// ---- edge scatter: s1[dst] += norm * x[src]  (one wave32 per edge, -------
// ---- 2 contiguous floats per lane -> 64 elems) ----------------------------
__global__ void ngcf_scatter_kernel(const float* __restrict__ xin, long xstride,
                                    const int* __restrict__ src,
                                    const int* __restrict__ dst,
                                    const float* __restrict__ nrm,
                                    float* __restrict__ s1, int n_edges) {
    long gid = (long)blockIdx.x * blockDim.x + threadIdx.x;
    int e = (int)(gid >> 5);
    if (e >= n_edges) return;
    int lane = (int)(gid & 31);
    int sj = src[e], di = dst[e];
    float w = nrm[e];
    const float* xr = xin + (long)sj * xstride + 2 * lane;
    float v0 = xr[0] * w;
    float v1 = xr[1] * w;
    float* sr = s1 + (long)di * DFEAT + 2 * lane;
    atomicAdd(sr, v0);
    atomicAdd(sr + 1, v1);
}

// ---- fused node GEMM:  out = act( (x+s1) @ W1 + (x.*s1) @ W2 ) ------------
// One wave32 computes one 16x16 tile of the [N x 64] result using
// V_WMMA_F32_16X16X4_F32.  A-layout: lane holds A[r][kk],A[r][kk+1] with
// r=lane%16, kk=k+2*(lane/16).  B-layout mirrored on columns.  D: acc[v] is
// D[v + 8*(lane/16)][lane%16].  Guards are wave-uniform (EXEC all-ones).
__global__ void ngcf_gemm_kernel(const float* __restrict__ xin, long xstride,
                                 const float* __restrict__ s1,
                                 const float* __restrict__ W1,
                                 const float* __restrict__ W2,
                                 float* __restrict__ out, long ostride,
                                 int n_nodes, int apply_act) {
    int wave = (blockIdx.x * blockDim.x + threadIdx.x) >> 5;
    int lane = threadIdx.x & 31;
    int row_tile = wave >> 2;        // 64/16 = 4 column tiles per row tile
    int col0 = (wave & 3) * 16;
    int row0 = row_tile * 16;
    if (row0 + 16 > n_nodes) return; // wave-uniform

    int r = lane & 15;
    int half = lane >> 4;            // 0 or 1

    const float* xrow = xin + (long)(row0 + r) * xstride;
    const float* srow = s1 + (long)(row0 + r) * DFEAT;

    v8f acc = {};
#pragma unroll
    for (int k = 0; k < DFEAT; k += 4) {
        int kk = k + 2 * half;
        float x0 = xrow[kk], x1 = xrow[kk + 1];
        float s0 = srow[kk], sv1 = srow[kk + 1];
        v2f a_sum = { x0 + s0, x1 + sv1 };      // (x + s1) row fragment
        v2f a_prd = { x0 * s0, x1 * sv1 };      // (x .* s1) row fragment
        v2f b1 = { W1[kk * DFEAT + col0 + r], W1[(kk + 1) * DFEAT + col0 + r] };
        v2f b2 = { W2[kk * DFEAT + col0 + r], W2[(kk + 1) * DFEAT + col0 + r] };
        acc = __builtin_amdgcn_wmma_f32_16x16x4_f32(
            false, a_sum, false, b1, (short)0, acc, false, false);
        acc = __builtin_amdgcn_wmma_f32_16x16x4_f32(
            false, a_prd, false, b2, (short)0, acc, false, false);
    }

#pragma unroll
    for (int v = 0; v < 8; ++v) {
        int m = v + 8 * half;
        float val = acc[v];
        if (apply_act) val = (val > 0.0f) ? val : NEG_SLOPE * val;
        out[(long)(row0 + m) * ostride + col0 + r] = val;
    }
}

extern "C" void kernel_launch(void* const* d_in, const int* in_sizes, int n_in,
                              void* d_out, int out_size, void* d_ws, size_t ws_size,
                              hipStream_t stream) {
    const float* x   = (const float*)d_in[0];
    const float* W1a = (const float*)d_in[1];
    const float* W2a = (const float*)d_in[2];
    const float* W1b = (const float*)d_in[3];
    const float* W2b = (const float*)d_in[4];
    const int*   src = (const int*)d_in[5];
    const int*   dst = (const int*)d_in[6];

    const int n_nodes = in_sizes[0] / DFEAT;   // 100000 (divisible by 16)
    const int n_edges = in_sizes[5];           // 3200000
    float* out = (float*)d_out;                // [N, 3*64] row-major

    // workspace: deg[N] | norm[E] | s1[N*64]
    float* deg  = (float*)d_ws;
    float* nrm  = deg + n_nodes;
    float* s1   = nrm + n_edges;

    const int T = 256;
    const int eblocks = (n_edges + T - 1) / T;

    hipMemsetAsync(deg, 0, (size_t)n_nodes * sizeof(float), stream);
    ngcf_deg_kernel<<<eblocks, T, 0, stream>>>(dst, deg, n_edges);
    ngcf_norm_kernel<<<eblocks, T, 0, stream>>>(src, dst, deg, nrm, n_edges);

    const long sc_threads = (long)n_edges * 32;
    const int  sc_blocks  = (int)((sc_threads + T - 1) / T);
    const int  gm_waves   = (n_nodes / 16) * (DFEAT / 16);
    const int  gm_blocks  = (gm_waves * 32 + T - 1) / T;

    struct Layer {
        const float* xin; long xstride;
        const float* W1; const float* W2;
        float* o; int act;
    };
    const long OS = 3 * DFEAT; // output row stride (192)
    Layer layers[3] = {
        { x,             DFEAT, W1a, W2a, out + 0 * DFEAT, 1 }, // h1
        { out + 0*DFEAT, OS,    W1b, W2b, out + 1 * DFEAT, 1 }, // h2 (reads h1)
        { out + 1*DFEAT, OS,    W1b, W2b, out + 2 * DFEAT, 0 }, // h3 (reads h2)
    };

    for (int l = 0; l < 3; ++l) {
        hipMemsetAsync(s1, 0, (size_t)n_nodes * DFEAT * sizeof(float), stream);
        ngcf_scatter_kernel<<<sc_blocks, T, 0, stream>>>(
            layers[l].xin, layers[l].xstride, src, dst, nrm, s1, n_edges);
        ngcf_gemm_kernel<<<gm_blocks, T, 0, stream>>>(
            layers[l].xin, layers[l].xstride, s1,
            layers[l].W1, layers[l].W2,
            layers[l].o, OS, n_nodes, layers[l].act);
    }
}